// GConvGRU_RecurrentGCN_16192026706531
// MI455X (gfx1250) — compile-verified
//
#include <hip/hip_runtime.h>
#include <math.h>

// GConvGRU (ChebConv K=2) single step with H0 = 0  ==>  only the Z and H~ gates
// on x survive:
//   tx1 = L_hat @ x,   L_hat = -D^-1/2 A D^-1/2
//   Z   = sigmoid(x@Wz0 + tx1@Wz1 + bz_x + bz_h)
//   Ht  = tanh   (x@Wh0 + tx1@Wh1 + bh_x + bh_h)
//   H   = (1-Z)*Ht;   out = relu(H) @ lin_W + lin_b
//
// Dense part done with V_WMMA_F32_16X16X4_F32 (full fp32 precision), one
// 16-node tile per wave32, effective K = 16 ([x|tx1]), N = 32 (2 col tiles).

typedef float v2f __attribute__((ext_vector_type(2)));
typedef float v8f __attribute__((ext_vector_type(8)));

#define CIN 8
#define FOUT 32

__global__ void zero_f32_kernel(float* __restrict__ p, int n) {
    int i = blockIdx.x * blockDim.x + threadIdx.x;
    if (i < n) p[i] = 0.0f;
}

// deg[src[e]] += ew[e]
__global__ void degree_kernel(const int* __restrict__ src,
                              const float* __restrict__ ew,
                              float* __restrict__ deg, int E) {
    int e = blockIdx.x * blockDim.x + threadIdx.x;
    if (e < E) atomicAdd(&deg[src[e]], ew[e]);
}

// deg -> D^{-1/2} in place (0 where deg<=0, matching the reference)
__global__ void dinv_kernel(float* __restrict__ deg, int N) {
    int i = blockIdx.x * blockDim.x + threadIdx.x;
    if (i < N) {
        float d = deg[i];
        deg[i] = (d > 0.0f) ? (1.0f / sqrtf(d)) : 0.0f;
    }
}

// tx1[dst, c] += (-dinv[src]*ew*dinv[dst]) * x[src, c]; 8 lanes per edge so the
// 8 atomic f32 adds per edge hit contiguous addresses (coalesced atomics).
__global__ void scatter_kernel(const float* __restrict__ x,
                               const int* __restrict__ src,
                               const int* __restrict__ dst,
                               const float* __restrict__ ew,
                               const float* __restrict__ dinv,
                               float* __restrict__ tx1, int E) {
    int t = blockIdx.x * blockDim.x + threadIdx.x;
    int e = t >> 3;
    int c = t & 7;
    if (e < E) {
        int s = src[e];
        int d = dst[e];
        float w = -dinv[s] * ew[e] * dinv[d];
        atomicAdd(&tx1[d * CIN + c], w * x[s * CIN + c]);
    }
}

__device__ __forceinline__ float sigmoidf_(float v) {
    return 1.0f / (1.0f + expf(-v));
}

// One wave32 handles 16 nodes. A_eff(16x16) = [x row | tx1 row], split into
// four K=4 chunks for V_WMMA_F32_16X16X4_F32. B_eff(16x32) is the stacked
// (2,8,32) weight tensor viewed flat as (16,32); two 16-column tiles.
__global__ void node_gate_kernel(const float* __restrict__ x,
                                 const float* __restrict__ tx1,
                                 const float* __restrict__ Wz,   // (2,8,32) flat = (16,32)
                                 const float* __restrict__ Wh,   // (2,8,32) flat = (16,32)
                                 const float* __restrict__ bz_x,
                                 const float* __restrict__ bz_h,
                                 const float* __restrict__ bh_x,
                                 const float* __restrict__ bh_h,
                                 const float* __restrict__ linW, // (32,)
                                 const float* __restrict__ linB, // (1,)
                                 float* __restrict__ out, int N) {
    const int lane = threadIdx.x & 31;
    const int wave = threadIdx.x >> 5;
    const int base = (blockIdx.x * 8 + wave) * 16;   // first node of this tile
    if (base >= N) return;                            // wave-uniform exit

    const int mrow = lane & 15;     // row for A loads / column within tile for B,C,D
    const int hs   = lane >> 4;     // half-wave select: K pair (0,1) vs (2,3)
    const int arow = base + mrow;   // node whose features this lane loads for A
    const bool av  = (arow < N);

    // ---- A operand: four K=4 chunks of [x(0..7) | tx1(0..7)] --------------
    v2f aF[4];
#pragma unroll
    for (int kc = 0; kc < 4; ++kc) {
        int k0 = kc * 4 + hs * 2;   // k0 even -> pair stays inside x or tx1
        float a0 = 0.0f, a1 = 0.0f;
        if (av) {
            const float* srcp = (k0 < CIN) ? (x + arow * CIN + k0)
                                           : (tx1 + arow * CIN + (k0 - CIN));
            a0 = srcp[0];
            a1 = srcp[1];
        }
        aF[kc].x = a0;
        aF[kc].y = a1;
    }

    // ---- WMMA accumulation: 2 gates x 2 col tiles x 4 K chunks ------------
    v8f accZ[2], accH[2];
#pragma unroll
    for (int t = 0; t < 2; ++t) { accZ[t] = (v8f)(0.0f); accH[t] = (v8f)(0.0f); }

#pragma unroll
    for (int t = 0; t < 2; ++t) {
        const int col = t * 16 + mrow;
#pragma unroll
        for (int kc = 0; kc < 4; ++kc) {
            const int k0 = kc * 4 + hs * 2;
            v2f bz, bh;
            bz.x = Wz[(k0 + 0) * FOUT + col];
            bz.y = Wz[(k0 + 1) * FOUT + col];
            bh.x = Wh[(k0 + 0) * FOUT + col];
            bh.y = Wh[(k0 + 1) * FOUT + col];
            accZ[t] = __builtin_amdgcn_wmma_f32_16x16x4_f32(
                false, aF[kc], false, bz, (short)0, accZ[t], false, false);
            accH[t] = __builtin_amdgcn_wmma_f32_16x16x4_f32(
                false, aF[kc], false, bh, (short)0, accH[t], false, false);
        }
    }

    // ---- Gates + partial output reduction ---------------------------------
    // Lane owns D rows r (hs==0) or r+8 (hs==1), r=0..7, at col f = 16t+mrow.
    float p[8];
#pragma unroll
    for (int r = 0; r < 8; ++r) p[r] = 0.0f;

#pragma unroll
    for (int t = 0; t < 2; ++t) {
        const int f = t * 16 + mrow;
        const float bZ = bz_x[f] + bz_h[f];
        const float bH = bh_x[f] + bh_h[f];
        const float lw = linW[f];
#pragma unroll
        for (int r = 0; r < 8; ++r) {
            float Z  = sigmoidf_(accZ[t][r] + bZ);
            float Ht = tanhf(accH[t][r] + bH);
            float H  = (1.0f - Z) * Ht;
            float Hr = (H > 0.0f) ? H : 0.0f;
            p[r] += Hr * lw;
        }
    }

    // Butterfly sum across the 16 lanes of each half-wave (cols 0..15):
    // xor masks 1,2,4,8 never cross the 16-lane half boundary.
#pragma unroll
    for (int m = 1; m < 16; m <<= 1) {
#pragma unroll
        for (int r = 0; r < 8; ++r) p[r] += __shfl_xor(p[r], m, 32);
    }

    if (mrow == 0) {
        const float lb = linB[0];
#pragma unroll
        for (int r = 0; r < 8; ++r) {
            int node = base + hs * 8 + r;
            if (node < N) out[node] = p[r] + lb;
        }
    }
}

extern "C" void kernel_launch(void* const* d_in, const int* in_sizes, int n_in,
                              void* d_out, int out_size, void* d_ws, size_t ws_size,
                              hipStream_t stream) {
    const float* x    = (const float*)d_in[0];
    const int*   eidx = (const int*)d_in[1];
    const float* ew   = (const float*)d_in[2];
    const float* Wz   = (const float*)d_in[3];   // Wz_x
    const float* bz_x = (const float*)d_in[4];
    const float* bz_h = (const float*)d_in[6];
    const float* Wh   = (const float*)d_in[11];  // Wh_x
    const float* bh_x = (const float*)d_in[12];
    const float* bh_h = (const float*)d_in[14];
    const float* linW = (const float*)d_in[15];
    const float* linB = (const float*)d_in[16];
    float* out = (float*)d_out;

    const int N = in_sizes[0] / CIN;
    const int E = in_sizes[2];
    const int* src = eidx;       // edge_index[0]
    const int* dst = eidx + E;   // edge_index[1]

    float* deg = (float*)d_ws;   // N floats
    float* tx1 = deg + N;        // N*CIN floats

    const int nz = N * (1 + CIN);
    zero_f32_kernel<<<(nz + 255) / 256, 256, 0, stream>>>(deg, nz);

    degree_kernel<<<(E + 255) / 256, 256, 0, stream>>>(src, ew, deg, E);

    dinv_kernel<<<(N + 255) / 256, 256, 0, stream>>>(deg, N);

    const long long st = (long long)E * CIN;
    scatter_kernel<<<(int)((st + 255) / 256), 256, 0, stream>>>(
        x, src, dst, ew, deg, tx1, E);

    // 8 waves/block * 16 nodes/wave = 128 nodes per block
    node_gate_kernel<<<(N + 127) / 128, 256, 0, stream>>>(
        x, tx1, Wz, Wh, bz_x, bz_h, bh_x, bh_h, linW, linB, out, N);
}